// decoder_61263413510322
// MI455X (gfx1250) — compile-verified
//
#include <hip/hip_runtime.h>
#include <hip/hip_bf16.h>

typedef __attribute__((ext_vector_type(2))) float v2f;
typedef __attribute__((ext_vector_type(8))) float v8f;

static constexpr int B   = 256;
static constexpr int L   = 400;
static constexpr int D   = 512;
static constexpr int H   = 512;
static constexpr int A   = 128;
static constexpr int ASP = 64;
static constexpr int V   = 30000;
static constexpr float P_DROP = 0.1f;
static constexpr float EPS = 1e-6f;

__device__ __forceinline__ float sigmoidf(float x) { return 1.0f / (1.0f + __expf(-x)); }

__device__ __forceinline__ unsigned hash_u32(unsigned x) {
    x ^= x >> 16; x *= 0x7feb352du;
    x ^= x >> 15; x *= 0x846ca68bu;
    x ^= x >> 16; return x;
}

// ---------------------------------------------------------------------------
// 1) dropout on x (deterministic hash RNG; JAX threefry not reproducible)
// ---------------------------------------------------------------------------
__global__ void dropout_kernel(const float* __restrict__ x, float* __restrict__ xd) {
    int i = blockIdx.x * blockDim.x + threadIdx.x;
    if (i >= B * D) return;
    unsigned r = hash_u32((unsigned)i ^ 0x2545F491u);
    float u = (float)(r & 0xFFFFFFu) * (1.0f / 16777216.0f);
    float keep = (u < (1.0f - P_DROP)) ? (1.0f / (1.0f - P_DROP)) : 0.0f;
    xd[i] = x[i] * keep;
}

// ---------------------------------------------------------------------------
// fp32 WMMA tile-group core: copy-free modulo-2 software pipeline.
// Step s consumes fragment buffer (s&1) with NT independent WMMAs, then
// refills that same buffer with the data for step s+2.  Pipeline distance 2
// => every load batch has ~2*NT WMMAs of latency cover, and there are no
// register-to-register copies to confuse the allocator.
// ---------------------------------------------------------------------------
template <int NT, int K>
__device__ __forceinline__ void wmma_tilegroup(const float* __restrict__ arow,
                                               const float* const (&brow)[NT],
                                               v8f (&c)[NT]) {
    constexpr int STEPS = K / 4;   // one WMMA group per 4 K-values
    static_assert(STEPS >= 2, "need at least two K-steps");

#pragma unroll
    for (int j = 0; j < NT; ++j) c[j] = (v8f){};

    v2f a[2];
    v2f b[2][NT];
    // prologue: fill both pipeline buffers (steps 0 and 1)
#pragma unroll
    for (int s = 0; s < 2; ++s) {
        const int k = 4 * s;
        a[s].x = arow[k]; a[s].y = arow[k + 1];
#pragma unroll
        for (int j = 0; j < NT; ++j) {
            b[s][j].x = brow[j][k];
            b[s][j].y = brow[j][k + 1];
        }
    }

#pragma unroll 2
    for (int s = 0; s < STEPS - 2; ++s) {
        const int buf = s & 1;
#pragma unroll
        for (int j = 0; j < NT; ++j)
            c[j] = __builtin_amdgcn_wmma_f32_16x16x4_f32(false, a[buf], false, b[buf][j],
                                                         (short)0, c[j], false, false);
        const int k = 4 * (s + 2);   // refill this buffer for step s+2
        a[buf].x = arow[k]; a[buf].y = arow[k + 1];
#pragma unroll
        for (int j = 0; j < NT; ++j) {
            b[buf][j].x = brow[j][k];
            b[buf][j].y = brow[j][k + 1];
        }
    }
    // epilogue: drain the last two buffered steps
#pragma unroll
    for (int s = STEPS - 2; s < STEPS; ++s) {
        const int buf = s & 1;
#pragma unroll
        for (int j = 0; j < NT; ++j)
            c[j] = __builtin_amdgcn_wmma_f32_16x16x4_f32(false, a[buf], false, b[buf][j],
                                                         (short)0, c[j], false, false);
    }
}

// ---------------------------------------------------------------------------
// 2) fp32 WMMA GEMM, register-blocked along N + pipelined.
//    C[M,N] = A[M,K] @ Bt[N,K]^T + bias[N].
// ---------------------------------------------------------------------------
template <int NT, int K>
__global__ void gemm_bias_wmma(const float* __restrict__ Am,   // M x K row-major
                               const float* __restrict__ Bt,   // N x K row-major
                               const float* __restrict__ bias, // N
                               float* __restrict__ C,          // M x N
                               int N) {
    const int n0 = blockIdx.x * (16 * NT);
    const int m0 = blockIdx.y * 16;
    const int lane = threadIdx.x;       // 0..31
    const int idx  = lane & 15;
    const int kh   = (lane >> 4) * 2;   // 0 or 2: K-half handled by this half-wave

    const float* arow = Am + (size_t)(m0 + idx) * K + kh;
    const float* brow[NT];
#pragma unroll
    for (int j = 0; j < NT; ++j)
        brow[j] = Bt + (size_t)(n0 + j * 16 + idx) * K + kh;

    v8f c[NT];
    wmma_tilegroup<NT, K>(arow, brow, c);

    const int mbase = m0 + ((lane >> 4) << 3); // +8 for upper half-wave
#pragma unroll
    for (int j = 0; j < NT; ++j) {
        const int n = n0 + j * 16 + (lane & 15);
        const float bv = bias[n];
#pragma unroll
        for (int r = 0; r < 8; ++r)
            C[(size_t)(mbase + r) * N + n] = c[j][r] + bv;
    }
}

// ---------------------------------------------------------------------------
// 3) GRU gate combine: h = (1-z)*n + z*h_prev
// ---------------------------------------------------------------------------
__global__ void gru_gate_kernel(const float* __restrict__ gi,  // B x 3H
                                const float* __restrict__ gh,  // B x 3H
                                const float* __restrict__ h0,  // B x H
                                float* __restrict__ hbuf,      // B x H (ws)
                                float* __restrict__ hout) {    // B x H (d_out slot)
    int i = blockIdx.x * blockDim.x + threadIdx.x;
    if (i >= B * H) return;
    int b = i / H, j = i % H;
    size_t base = (size_t)b * (3 * H);
    float ir = gi[base + j],          hr = gh[base + j];
    float iz = gi[base + H + j],      hz = gh[base + H + j];
    float in_ = gi[base + 2 * H + j], hn = gh[base + 2 * H + j];
    float r = sigmoidf(ir + hr);
    float z = sigmoidf(iz + hz);
    float n = tanhf(in_ + r * hn);
    float h = (1.0f - z) * n + z * h0[i];
    hbuf[i] = h;
    hout[i] = h;
}

// ---------------------------------------------------------------------------
// 4) additive attention over encoder_outputs; one block per batch element.
//    Bandwidth-bound (2 passes over 210MB); no matmul structure.
// ---------------------------------------------------------------------------
__global__ void attn1_kernel(const float* __restrict__ enc,     // L x B x H
                             const int*   __restrict__ summary, // L x B
                             const float* __restrict__ h,       // B x H
                             const float* __restrict__ W8,      // 1 x 2H
                             const float* __restrict__ b8,      // 1
                             float* __restrict__ a1out) {       // B x H (d_out slot)
    const int b = blockIdx.x;
    const int t = threadIdx.x;  // 256 threads
    __shared__ float sc[L];
    __shared__ float red[256];
    __shared__ float hd, ssum;

    // hd = dot(h[b], w_h) + b8
    float p = h[(size_t)b * H + t] * W8[H + t] +
              h[(size_t)b * H + t + 256] * W8[H + t + 256];
    red[t] = p;
    __syncthreads();
    for (int s = 128; s > 0; s >>= 1) {
        if (t < s) red[t] += red[t + s];
        __syncthreads();
    }
    if (t == 0) hd = red[0] + b8[0];
    __syncthreads();

    // scores -> exp(tanh(.)) * mask
    for (int l = t; l < L; l += 256) {
        const float* e = enc + ((size_t)l * B + b) * H;
        float d = 0.0f;
        for (int k = 0; k < H; ++k) d += e[k] * W8[k];
        float m = (summary[(size_t)l * B + b] > 0) ? 1.0f : 0.0f;
        sc[l] = __expf(tanhf(d + hd)) * m;
    }
    __syncthreads();

    // sum over l
    float q = sc[t] + ((t + 256 < L) ? sc[t + 256] : 0.0f);
    red[t] = q;
    __syncthreads();
    for (int s = 128; s > 0; s >>= 1) {
        if (t < s) red[t] += red[t + s];
        __syncthreads();
    }
    if (t == 0) ssum = red[0];
    __syncthreads();
    const float inv = 1.0f / (ssum + EPS);

    // weighted sum over l for each feature f
    for (int f = t; f < H; f += 256) {
        float acc = 0.0f;
        for (int l = 0; l < L; ++l)
            acc += sc[l] * enc[((size_t)l * B + b) * H + f];
        a1out[(size_t)b * H + f] = acc * inv;
    }
}

// ---------------------------------------------------------------------------
// 5) attention over {gamma_u, gamma_i}; one block (128 threads) per batch.
// ---------------------------------------------------------------------------
__global__ void attn2_kernel(const float* __restrict__ gamma_u, // B x A
                             const float* __restrict__ gamma_i, // B x A
                             const float* __restrict__ h,       // B x H
                             const float* __restrict__ W10,     // 1 x (A+H)
                             const float* __restrict__ b10,     // 1
                             float* __restrict__ a2out) {       // B x A (d_out slot)
    const int b = blockIdx.x;
    const int t = threadIdx.x;  // 128 threads
    __shared__ float red[128];
    __shared__ float hd, e0, e1;

    float p = 0.0f;
#pragma unroll
    for (int i = 0; i < 4; ++i)
        p += h[(size_t)b * H + t + i * 128] * W10[A + t + i * 128];
    red[t] = p;
    __syncthreads();
    for (int s = 64; s > 0; s >>= 1) {
        if (t < s) red[t] += red[t + s];
        __syncthreads();
    }
    if (t == 0) hd = red[0] + b10[0];
    __syncthreads();

    red[t] = gamma_u[(size_t)b * A + t] * W10[t];
    __syncthreads();
    for (int s = 64; s > 0; s >>= 1) {
        if (t < s) red[t] += red[t + s];
        __syncthreads();
    }
    if (t == 0) e0 = __expf(tanhf(red[0] + hd));
    __syncthreads();

    red[t] = gamma_i[(size_t)b * A + t] * W10[t];
    __syncthreads();
    for (int s = 64; s > 0; s >>= 1) {
        if (t < s) red[t] += red[t + s];
        __syncthreads();
    }
    if (t == 0) e1 = __expf(tanhf(red[0] + hd));
    __syncthreads();

    const float inv = 1.0f / (e0 + e1 + EPS);
    a2out[(size_t)b * A + t] =
        (e0 * gamma_u[(size_t)b * A + t] + e1 * gamma_i[(size_t)b * A + t]) * inv;
}

// ---------------------------------------------------------------------------
// 6) s_ui + a_3_t; one block (64 threads) per batch element.
// ---------------------------------------------------------------------------
__global__ void a3_kernel(const float* __restrict__ beta_u, // B x ASP
                          const float* __restrict__ beta_i, // B x ASP
                          const float* __restrict__ xd,     // B x D
                          const float* __restrict__ h,      // B x H
                          const float* __restrict__ W11,    // ASP x 2*ASP
                          const float* __restrict__ b11,    // ASP
                          const float* __restrict__ W12,    // ASP x (ASP+D+H)
                          const float* __restrict__ b12,    // ASP
                          float* __restrict__ a3) {         // B x ASP (ws)
    const int b = blockIdx.x;
    const int j = threadIdx.x; // 64 threads
    __shared__ float sui[ASP];

    float acc = b11[j];
    const float* w11 = W11 + (size_t)j * (2 * ASP);
    for (int k = 0; k < ASP; ++k) acc += beta_u[(size_t)b * ASP + k] * w11[k];
    for (int k = 0; k < ASP; ++k) acc += beta_i[(size_t)b * ASP + k] * w11[ASP + k];
    sui[j] = acc;
    __syncthreads();

    const int KC = ASP + D + H; // 1088
    const float* w12 = W12 + (size_t)j * KC;
    float a = b12[j];
    for (int k = 0; k < ASP; ++k) a += sui[k] * w12[k];
    for (int k = 0; k < D;   ++k) a += xd[(size_t)b * D + k] * w12[ASP + k];
    for (int k = 0; k < H;   ++k) a += h[(size_t)b * H + k] * w12[ASP + D + k];
    a3[(size_t)b * ASP + j] = a;
}

// ---------------------------------------------------------------------------
// 7) word_probit GEMM (B x V, K=H) via fp32 WMMA, NT=5 N-blocking + pipelined,
//    fused aspect-probit epilogue.  1875 N-tiles / 5 = 375 blocks in x.
// ---------------------------------------------------------------------------
template <int NT>
__global__ void wordprobit_wmma(const float* __restrict__ h,       // B x H
                                const float* __restrict__ Wx,      // V x H
                                const float* __restrict__ bx,      // V
                                const float* __restrict__ a3,      // B x ASP
                                const int*   __restrict__ raspect, // V
                                const float* __restrict__ rmask,   // B x V
                                float* __restrict__ out) {         // B x V
    const int n0 = blockIdx.x * (16 * NT); // vocab tile group
    const int m0 = blockIdx.y * 16;        // batch tile
    const int lane = threadIdx.x;
    const int idx  = lane & 15;
    const int kh   = (lane >> 4) * 2;

    const float* arow = h + (size_t)(m0 + idx) * H + kh;
    const float* brow[NT];
#pragma unroll
    for (int j = 0; j < NT; ++j) {
        brow[j] = Wx + (size_t)(n0 + j * 16 + idx) * H + kh;
        __builtin_prefetch(brow[j], 0, 1);   // global_prefetch_b8 on the Wx stream
    }

    v8f c[NT];
    wmma_tilegroup<NT, H>(arow, brow, c);

    const int mbase = m0 + ((lane >> 4) << 3);
#pragma unroll
    for (int j = 0; j < NT; ++j) {
        const int n = n0 + j * 16 + (lane & 15);
        const float bv = bx[n];
        const int asp = raspect[n];
#pragma unroll
        for (int r = 0; r < 8; ++r) {
            const int m = mbase + r;
            out[(size_t)m * V + n] =
                c[j][r] + bv + a3[(size_t)m * ASP + asp] * rmask[(size_t)m * V + n];
        }
    }
}

// ---------------------------------------------------------------------------
extern "C" void kernel_launch(void* const* d_in, const int* in_sizes, int n_in,
                              void* d_out, int out_size, void* d_ws, size_t ws_size,
                              hipStream_t stream) {
    const float* x        = (const float*)d_in[0];
    const float* h0       = (const float*)d_in[1];
    const float* enc      = (const float*)d_in[2];
    const int*   summary  = (const int*)d_in[3];
    const float* gamma_u  = (const float*)d_in[4];
    const float* gamma_i  = (const float*)d_in[5];
    const float* beta_u   = (const float*)d_in[6];
    const float* beta_i   = (const float*)d_in[7];
    const int*   raspect  = (const int*)d_in[8];
    const float* rmask    = (const float*)d_in[9];
    const float* W_ih     = (const float*)d_in[10];
    const float* W_hh     = (const float*)d_in[11];
    const float* b_ih     = (const float*)d_in[12];
    const float* b_hh     = (const float*)d_in[13];
    const float* W8       = (const float*)d_in[14];
    const float* b8       = (const float*)d_in[15];
    const float* W10      = (const float*)d_in[16];
    const float* b10      = (const float*)d_in[17];
    const float* W11      = (const float*)d_in[18];
    const float* b11      = (const float*)d_in[19];
    const float* W12      = (const float*)d_in[20];
    const float* b12      = (const float*)d_in[21];
    const float* Wx       = (const float*)d_in[22];
    const float* bx       = (const float*)d_in[23];

    float* out = (float*)d_out;
    // d_out flat layout: [B*V][B*H (h)][B*H (a1)][B*A (a2)]
    float* out_probit = out;
    float* out_h      = out + (size_t)B * V;
    float* out_a1     = out_h + (size_t)B * H;
    float* out_a2     = out_a1 + (size_t)B * H;

    // workspace carve-out
    float* ws   = (float*)d_ws;
    float* xd   = ws;                       // B*D
    float* gi   = xd + (size_t)B * D;       // B*3H
    float* gh   = gi + (size_t)B * 3 * H;   // B*3H
    float* hbuf = gh + (size_t)B * 3 * H;   // B*H
    float* a3   = hbuf + (size_t)B * H;     // B*ASP

    // 1) dropout
    dropout_kernel<<<(B * D + 255) / 256, 256, 0, stream>>>(x, xd);

    // 2) GRU input/hidden GEMMs (WMMA fp32, NT=4: 96 N-tiles / 4 = 24)
    gemm_bias_wmma<4, D><<<dim3(3 * H / 64, B / 16), 32, 0, stream>>>(xd, W_ih, b_ih, gi, 3 * H);
    gemm_bias_wmma<4, H><<<dim3(3 * H / 64, B / 16), 32, 0, stream>>>(h0, W_hh, b_hh, gh, 3 * H);

    // 3) gate combine -> h (ws + output slot)
    gru_gate_kernel<<<(B * H + 255) / 256, 256, 0, stream>>>(gi, gh, h0, hbuf, out_h);

    // 4) encoder attention -> a_1_t
    attn1_kernel<<<B, 256, 0, stream>>>(enc, summary, hbuf, W8, b8, out_a1);

    // 5) gamma attention -> a_2_t
    attn2_kernel<<<B, 128, 0, stream>>>(gamma_u, gamma_i, hbuf, W10, b10, out_a2);

    // 6) s_ui + a_3_t
    a3_kernel<<<B, ASP, 0, stream>>>(beta_u, beta_i, xd, hbuf, W11, b11, W12, b12, a3);

    // 7) word probit + aspect probit (WMMA fp32, NT=5: 1875 N-tiles / 5 = 375)
    wordprobit_wmma<5><<<dim3(V / 80, B / 16), 32, 0, stream>>>(hbuf, Wx, bx, a3, raspect, rmask, out_probit);
}